// APPNPNet_48541720379898
// MI455X (gfx1250) — compile-verified
//
#include <hip/hip_runtime.h>
#include <hip/hip_bf16.h>

typedef __attribute__((ext_vector_type(2))) float v2f;
typedef __attribute__((ext_vector_type(8))) float v8f;

#define EMB_DIM 128
#define HIDDEN  256
#define OUT_C   64
#define K_ITERS 10
#define ALPHA   0.1f

// ---------------------------------------------------------------------------
// GCN normalization helpers
// ---------------------------------------------------------------------------
__global__ void deg_init_kernel(float* deg, int n) {
    int i = blockIdx.x * blockDim.x + threadIdx.x;
    if (i < n) deg[i] = 1.0f;   // self-loop contributes 1 to every node's degree
}

__global__ void deg_count_kernel(const int* __restrict__ dst, float* deg, int e) {
    int i = blockIdx.x * blockDim.x + threadIdx.x;
    if (i < e) atomicAdd(&deg[dst[i]], 1.0f);
}

__global__ void deg_rsqrt_kernel(float* deg, int n) {
    int i = blockIdx.x * blockDim.x + threadIdx.x;
    if (i < n) {
        float d = deg[i];
        deg[i] = (d > 0.0f) ? rsqrtf(d) : 0.0f;   // deg buffer now holds dis[]
    }
}

__global__ void edge_norm_kernel(const int* __restrict__ src, const int* __restrict__ dst,
                                 const float* __restrict__ dis, float* __restrict__ norm, int e) {
    int i = blockIdx.x * blockDim.x + threadIdx.x;
    if (i < e) norm[i] = dis[src[i]] * dis[dst[i]];
}

// ---------------------------------------------------------------------------
// Fused 2-layer MLP via native fp32 WMMA (V_WMMA_F32_16X16X4_F32)
// One block = 16 output rows, 4 waves. Wave w owns hidden cols [64w,64w+64)
// for GEMM1 and output cols [16w,16w+16) for GEMM2. Hidden tile lives in LDS.
// ---------------------------------------------------------------------------
__launch_bounds__(128)
__global__ void mlp_wmma_kernel(const int* __restrict__ xidx,
                                const float* __restrict__ emb,
                                const float* __restrict__ W1, const float* __restrict__ b1,
                                const float* __restrict__ W2, const float* __restrict__ b2,
                                float* __restrict__ h_out, float* __restrict__ x0_out,
                                int n) {
    const int tid  = threadIdx.x;       // 0..127
    const int wave = tid >> 5;          // 0..3
    const int lane = tid & 31;
    const int m    = lane & 15;         // row (A) / col (B,C) index within 16x16 tile
    const int g    = lane >> 4;         // K-pair group (VGPR pair holds K=2g, 2g+1)
    const int row0 = blockIdx.x * 16;

    __shared__ float sA[16][EMB_DIM + 2];   // gathered emb rows (pad: avoid bank aliasing)
    __shared__ float sH[16][HIDDEN + 2];    // relu'd hidden tile

    // Gather 16 embedding rows into LDS (coalesced across 128 threads).
    for (int i = tid; i < 16 * EMB_DIM; i += 128) {
        int r  = i >> 7;          // /EMB_DIM
        int c  = i & (EMB_DIM - 1);
        int rr = row0 + r;
        float v = 0.0f;
        if (rr < n) {
            int node = xidx[rr];
            v = emb[(size_t)node * EMB_DIM + c];
        }
        sA[r][c] = v;
    }
    __syncthreads();

    // -------- GEMM1: hidden[16 x 256] slice, K = 128 (32 wmma k-steps/tile) --------
    for (int t = 0; t < 4; ++t) {
        const int nbase = wave * 64 + t * 16;
        v8f acc = {0.f, 0.f, 0.f, 0.f, 0.f, 0.f, 0.f, 0.f};
        for (int s = 0; s < 32; ++s) {
            const int k = 4 * s + 2 * g;
            v2f a; a.x = sA[m][k];                    a.y = sA[m][k + 1];
            v2f b; b.x = W1[(size_t)k * HIDDEN + nbase + m];
                   b.y = W1[(size_t)(k + 1) * HIDDEN + nbase + m];
            acc = __builtin_amdgcn_wmma_f32_16x16x4_f32(
                false, a, false, b, (short)0, acc, false, false);
        }
        const float bias = b1[nbase + m];
        #pragma unroll
        for (int v = 0; v < 8; ++v) {
            float val = acc[v] + bias;
            val = val > 0.0f ? val : 0.0f;            // relu
            sH[v + 8 * g][nbase + m] = val;
        }
    }
    __syncthreads();

    // -------- GEMM2: out[16 x 16] tile per wave, K = 256 (64 wmma k-steps) --------
    {
        const int nbase = wave * 16;
        v8f acc = {0.f, 0.f, 0.f, 0.f, 0.f, 0.f, 0.f, 0.f};
        for (int s = 0; s < 64; ++s) {
            const int k = 4 * s + 2 * g;
            v2f a; a.x = sH[m][k];                    a.y = sH[m][k + 1];
            v2f b; b.x = W2[(size_t)k * OUT_C + nbase + m];
                   b.y = W2[(size_t)(k + 1) * OUT_C + nbase + m];
            acc = __builtin_amdgcn_wmma_f32_16x16x4_f32(
                false, a, false, b, (short)0, acc, false, false);
        }
        const float bias = b2[nbase + m];
        #pragma unroll
        for (int v = 0; v < 8; ++v) {
            const int row = row0 + v + 8 * g;
            if (row < n) {
                const float val = acc[v] + bias;
                const size_t idx = (size_t)row * OUT_C + nbase + m;
                h_out[idx]  = val;   // teleport term h
                x0_out[idx] = val;   // propagation state x0
            }
        }
    }
}

// ---------------------------------------------------------------------------
// Propagation: scatter-add of x[src]*norm into agg[dst].
// 16 threads/edge, each does one float4 (b128) gather + 4 f32 atomics.
// The 16 lanes of an edge read one contiguous 256B row (L2-resident).
// ---------------------------------------------------------------------------
__global__ void scatter_kernel(const int* __restrict__ src, const int* __restrict__ dst,
                               const float* __restrict__ norm,
                               const float* __restrict__ x, float* __restrict__ agg,
                               int e) {
    const unsigned gid = blockIdx.x * blockDim.x + threadIdx.x;
    const unsigned ei  = gid >> 4;          // edge index
    const unsigned q   = gid & 15;          // float4 slot within the 64-ch row
    if (ei < (unsigned)e) {
        const int   s  = src[ei];
        const int   d  = dst[ei];
        const float nv = norm[ei];
        const float4 v = ((const float4*)(x + (size_t)s * OUT_C))[q];
        float* ap = agg + (size_t)d * OUT_C + q * 4;
        atomicAdd(ap + 0, v.x * nv);
        atomicAdd(ap + 1, v.y * nv);
        atomicAdd(ap + 2, v.z * nv);
        atomicAdd(ap + 3, v.w * nv);
    }
}

// x_new = (1-a)*(agg + dis[i]^2 * x_old) + a*h   (self-loop folded in; in-place
// on the agg buffer). float4-vectorized: 16 float4 per node.
__global__ void combine_kernel(float4* __restrict__ xnew_agg, const float4* __restrict__ xold,
                               const float* __restrict__ dis, const float4* __restrict__ h,
                               int total4) {
    const int i = blockIdx.x * blockDim.x + threadIdx.x;
    if (i < total4) {
        const int   node = i >> 4;          // 16 float4 per node (OUT_C/4)
        const float di   = dis[node];
        const float d2   = di * di;
        const float4 a  = xnew_agg[i];
        const float4 xo = xold[i];
        const float4 hh = h[i];
        float4 r;
        r.x = (1.0f - ALPHA) * (a.x + d2 * xo.x) + ALPHA * hh.x;
        r.y = (1.0f - ALPHA) * (a.y + d2 * xo.y) + ALPHA * hh.y;
        r.z = (1.0f - ALPHA) * (a.z + d2 * xo.z) + ALPHA * hh.z;
        r.w = (1.0f - ALPHA) * (a.w + d2 * xo.w) + ALPHA * hh.w;
        xnew_agg[i] = r;
    }
}

// ---------------------------------------------------------------------------
extern "C" void kernel_launch(void* const* d_in, const int* in_sizes, int n_in,
                              void* d_out, int out_size, void* d_ws, size_t ws_size,
                              hipStream_t stream) {
    (void)n_in; (void)out_size; (void)ws_size;

    const int*   xidx = (const int*)d_in[0];
    const int*   ei   = (const int*)d_in[1];
    const float* emb  = (const float*)d_in[2];
    const float* W1   = (const float*)d_in[3];
    const float* b1   = (const float*)d_in[4];
    const float* W2   = (const float*)d_in[5];
    const float* b2   = (const float*)d_in[6];
    float* out = (float*)d_out;

    const int N = in_sizes[0];
    const int E = in_sizes[1] / 2;
    const size_t NC = (size_t)N * OUT_C;

    // workspace layout: h | xping | dis | norm
    float* h     = (float*)d_ws;
    float* xping = h + NC;
    float* dis   = xping + NC;
    float* norm  = dis + N;

    const int* src = ei;       // ei[0]
    const int* dst = ei + E;   // ei[1]

    // --- GCN normalization ---
    deg_init_kernel <<<(N + 255) / 256, 256, 0, stream>>>(dis, N);
    deg_count_kernel<<<(E + 255) / 256, 256, 0, stream>>>(dst, dis, E);
    deg_rsqrt_kernel<<<(N + 255) / 256, 256, 0, stream>>>(dis, N);
    edge_norm_kernel<<<(E + 255) / 256, 256, 0, stream>>>(src, dst, dis, norm, E);

    // --- fused MLP (WMMA) -> h and x0 (x0 lives in d_out; K is even so the
    //     ping-pong ends back in d_out) ---
    mlp_wmma_kernel<<<(N + 15) / 16, 128, 0, stream>>>(xidx, emb, W1, b1, W2, b2,
                                                       h, out, N);

    // --- APPNP propagation, K rounds ---
    float* xa = out;     // current x
    float* xb = xping;   // next x (also serves as agg accumulator)
    const long long sthreads = (long long)E * 16;         // 16 threads/edge
    const int sblocks = (int)((sthreads + 255) / 256);
    const int total4  = (int)(NC / 4);
    const int cblocks = (total4 + 255) / 256;
    for (int k = 0; k < K_ITERS; ++k) {
        hipMemsetAsync(xb, 0, NC * sizeof(float), stream);
        scatter_kernel<<<sblocks, 256, 0, stream>>>(src, dst, norm, xa, xb, E);
        combine_kernel<<<cblocks, 256, 0, stream>>>((float4*)xb, (const float4*)xa,
                                                    dis, (const float4*)h, total4);
        float* t = xa; xa = xb; xb = t;
    }
    if (xa != out)  // K even -> dead path, kept for safety / determinism
        hipMemcpyAsync(out, xa, NC * sizeof(float), hipMemcpyDeviceToDevice, stream);
}